// NormalGMM_26740466385350
// MI455X (gfx1250) — compile-verified
//
#include <hip/hip_runtime.h>
#include <math.h>

#define EPSF 1e-10f
#define NB 32
#define NK 4
#define NM 3
#define NPIX 65536          // 256*256
#define TILE 256
#define LDSTRIDE 260        // TILE + 4 dword pad -> conflict-free row gathers
#define TILES_PER_B 256     // NPIX / TILE
#define STAT_WGS_PER_B 32
#define LL_WGS_PER_B 64

typedef float v2f __attribute__((ext_vector_type(2)));
typedef float v8f __attribute__((ext_vector_type(8)));

// Workspace layout (floats):
//   stats : [B][5][8]  rows 0..3 = component k, row 4 = mask; cols 0..6 = {S0, S1m, S2m}
//   params: [B][K][8]  {c0, mu0..2, inv2v0..2, pad}
//   ll    : [B]
#define WS_STATS  0
#define WS_PARAMS (NB * 5 * 8)                    // 1280
#define WS_LL     (WS_PARAMS + NB * NK * 8)       // 2304
#define WS_TOTAL  (WS_LL + NB)                    // 2336

__global__ __launch_bounds__(256) void gmm_zero(float* __restrict__ ws) {
    int i = blockIdx.x * 256 + threadIdx.x;
    if (i < WS_TOTAL) ws[i] = 0.0f;
}

// ---------------------------------------------------------------------------
// Pass 1: per-(b,k) sufficient statistics via V_WMMA_F32_16X16X4_F32.
// D[r][c] += sum over pixels of A_row_r(pix) * B_col_c(pix), with
//   A rows (s_a): 0..3 = mask*pred_k, 4 = mask, 5 = zeros
//   B cols (s_b): 0 = ones, 1..3 = x_m, 4..6 = x_m^2, 7 = zeros
// Operand gather is branch-free: each lane resolves one LDS base pointer per
// table (row select via min), then the unrolled loop is pure
// ds_load_b64 + wmma with immediate offsets. f32 operand layout (ISA 7.12.2):
// lane L, VGPR j holds pixel (group + 2*(L>>4) + j).
// ---------------------------------------------------------------------------
__global__ __launch_bounds__(256)
void gmm_stats(const float* __restrict__ pred, const float* __restrict__ inp,
               const int* __restrict__ heart, float* __restrict__ ws)
{
    __shared__ __align__(16) float s_a[6][LDSTRIDE];
    __shared__ __align__(16) float s_b[8][LDSTRIDE];
    __shared__ float s_red[8][5][16];

    const int b    = blockIdx.y;
    const int tid  = threadIdx.x;
    const int lane = tid & 31;
    const int wave = tid >> 5;

    const float* predb = pred  + (size_t)b * NK * NPIX;
    const float* inpb  = inp   + (size_t)b * NM * NPIX;
    const int*   hb    = heart + (size_t)b * NPIX;

    // Constant rows, written once (visible after first loop barrier).
    s_a[5][tid] = 0.0f;     // zero row for A rows 5..15
    s_b[0][tid] = 1.0f;     // ones column (c==0 -> S0 accumulation)
    s_b[7][tid] = 0.0f;     // zero column for c 7..15

    // Per-lane operand base pointers (uniform across the whole kernel).
    const int row   = lane & 15;             // A row / B column index
    const int half  = lane >> 4;             // K-dim half select
    const int a_idx = (row < 5) ? row : 5;
    const int b_idx = (row < 7) ? row : 7;
    const float* __restrict__ ap = &s_a[a_idx][wave * 32 + 2 * half];
    const float* __restrict__ bp = &s_b[b_idx][wave * 32 + 2 * half];

    v8f acc = {0.f, 0.f, 0.f, 0.f, 0.f, 0.f, 0.f, 0.f};

    for (int t = blockIdx.x; t < TILES_PER_B; t += gridDim.x) {
        const int p0 = t * TILE;
        __syncthreads();   // previous tile fully consumed / init rows visible
        {
            const float mk = (hb[p0 + tid] == 1) ? 1.0f : 0.0f;
            s_a[4][tid] = mk;
#pragma unroll
            for (int k = 0; k < NK; ++k)
                s_a[k][tid] = predb[k * NPIX + p0 + tid] * mk;
#pragma unroll
            for (int m = 0; m < NM; ++m) {
                const float x = inpb[m * NPIX + p0 + tid];
                s_b[1 + m][tid] = x;
                s_b[4 + m][tid] = x * x;
            }
        }
        __syncthreads();

        // 8 groups of 4 pixels: 2x ds_load_b64 + 1x wmma each, no VALU.
#pragma unroll
        for (int j = 0; j < 8; ++j) {
            v2f av = *(const v2f*)(ap + j * 4);
            v2f bv = *(const v2f*)(bp + j * 4);
            acc = __builtin_amdgcn_wmma_f32_16x16x4_f32(
                false, av, false, bv, (short)0, acc, false, false);
        }
    }

    // D layout: lanes 0-15 hold columns N=lane, VGPR r holds row M=r.
    __syncthreads();
    if (lane < 16) {
#pragma unroll
        for (int r = 0; r < 5; ++r) s_red[wave][r][lane] = acc[r];
    }
    __syncthreads();
    if (tid < 5 * 16) {
        const int r = tid >> 4, c = tid & 15;
        float sum = 0.0f;
#pragma unroll
        for (int w = 0; w < 8; ++w) sum += s_red[w][r][c];
        if (c < 7)
            atomicAdd(&ws[WS_STATS + (b * 5 + r) * 8 + c], sum);
    }
}

// ---------------------------------------------------------------------------
// Pass 2: 128 threads, one per (b,k): alpha, mu, var -> folded params.
// ---------------------------------------------------------------------------
__global__ __launch_bounds__(128) void gmm_finalize(float* __restrict__ ws)
{
    const int t = threadIdx.x;
    if (t >= NB * NK) return;
    const int b = t >> 2, k = t & 3;

    const float* st = ws + WS_STATS + (b * 5 + k) * 8;
    const float  n  = ws[WS_STATS + (b * 5 + 4) * 8 + 0];
    const float  S0 = st[0];
    const float  s  = S0 + EPSF;
    const float  alpha = S0 / n;

    float c0 = __logf(alpha);
    float* pp = ws + WS_PARAMS + (b * NK + k) * 8;
#pragma unroll
    for (int m = 0; m < NM; ++m) {
        const float S1 = st[1 + m], S2 = st[4 + m];
        const float mu = S1 / s;
        const float var = (S2 - 2.0f * mu * S1 + mu * mu * S0) / s + EPSF;
        c0 += -0.5f * __logf(6.283185307179586f * var);
        pp[1 + m] = mu;
        pp[4 + m] = 0.5f / var;    // 1/(2v)
    }
    pp[0] = c0;
    pp[7] = 0.0f;
}

// ---------------------------------------------------------------------------
// Pass 3: per-pixel mixture log-likelihood, block-reduced, atomic per batch.
// Only reads inputs+heart (32 MB) -> L2-resident after pass 1.
// ---------------------------------------------------------------------------
__global__ __launch_bounds__(256)
void gmm_ll(const float* __restrict__ inp, const int* __restrict__ heart,
            float* __restrict__ ws)
{
    __shared__ float s_par[NK][8];
    __shared__ float s_red[256];

    const int b = blockIdx.y, tid = threadIdx.x;
    if (tid < NK * 8) s_par[tid >> 3][tid & 7] = ws[WS_PARAMS + b * NK * 8 + tid];
    __syncthreads();

    const float* inpb = inp + (size_t)b * NM * NPIX;
    const int*   hb   = heart + (size_t)b * NPIX;

    float lsum = 0.0f;
    for (int t = blockIdx.x; t < TILES_PER_B; t += gridDim.x) {
        const int p  = t * TILE + tid;
        const float mk = (hb[p] == 1) ? 1.0f : 0.0f;
        const float x0 = inpb[p];
        const float x1 = inpb[NPIX + p];
        const float x2 = inpb[2 * NPIX + p];
        float mix = 0.0f;
#pragma unroll
        for (int k = 0; k < NK; ++k) {
            const float d0 = x0 - s_par[k][1];
            const float d1 = x1 - s_par[k][2];
            const float d2 = x2 - s_par[k][3];
            const float e  = s_par[k][0]
                           - (d0 * d0 * s_par[k][4] + d1 * d1 * s_par[k][5] + d2 * d2 * s_par[k][6]);
            mix += __expf(e);
        }
        lsum += mk * __logf(mix + EPSF);
    }

    s_red[tid] = lsum;
    __syncthreads();
#pragma unroll
    for (int sft = 128; sft > 0; sft >>= 1) {
        if (tid < sft) s_red[tid] += s_red[tid + sft];
        __syncthreads();
    }
    if (tid == 0) atomicAdd(&ws[WS_LL + b], s_red[0]);
}

// ---------------------------------------------------------------------------
// Pass 4: mean over batches of (-sum_ll / n) -> scalar output.
// ---------------------------------------------------------------------------
__global__ __launch_bounds__(32)
void gmm_out(const float* __restrict__ ws, float* __restrict__ out)
{
    __shared__ float s[32];
    const int t = threadIdx.x;
    const float n = ws[WS_STATS + (t * 5 + 4) * 8 + 0];
    s[t] = -ws[WS_LL + t] / n;
    __syncthreads();
#pragma unroll
    for (int k = 16; k > 0; k >>= 1) {
        if (t < k) s[t] += s[t + k];
        __syncthreads();
    }
    if (t == 0) out[0] = s[0] / (float)NB;
}

extern "C" void kernel_launch(void* const* d_in, const int* in_sizes, int n_in,
                              void* d_out, int out_size, void* d_ws, size_t ws_size,
                              hipStream_t stream) {
    const float* pred  = (const float*)d_in[0];
    const float* inp   = (const float*)d_in[1];
    const int*   heart = (const int*)d_in[2];
    float* ws  = (float*)d_ws;
    float* out = (float*)d_out;

    gmm_zero<<<(WS_TOTAL + 255) / 256, 256, 0, stream>>>(ws);
    gmm_stats<<<dim3(STAT_WGS_PER_B, NB), 256, 0, stream>>>(pred, inp, heart, ws);
    gmm_finalize<<<1, 128, 0, stream>>>(ws);
    gmm_ll<<<dim3(LL_WGS_PER_B, NB), 256, 0, stream>>>(inp, heart, ws);
    gmm_out<<<1, 32, 0, stream>>>(ws, out);
}